// ChebNet_40604620816841
// MI455X (gfx1250) — compile-verified
//
#include <hip/hip_runtime.h>
#include <hip/hip_bf16.h>

typedef __attribute__((ext_vector_type(2))) float v2f;
typedef __attribute__((ext_vector_type(8))) float v8f;

// ---------------------------------------------------------------------------
// zero fill (grid-stride)
// ---------------------------------------------------------------------------
__global__ void zero_kernel(float* __restrict__ p, size_t n) {
    size_t i = (size_t)blockIdx.x * blockDim.x + threadIdx.x;
    size_t stride = (size_t)gridDim.x * blockDim.x;
    for (; i < n; i += stride) p[i] = 0.0f;
}

// ---------------------------------------------------------------------------
// deg[row[e]] += w[e]
// ---------------------------------------------------------------------------
__global__ void deg_kernel(const int* __restrict__ row, const float* __restrict__ w,
                           float* __restrict__ deg, int E) {
    int e = blockIdx.x * blockDim.x + threadIdx.x;
    if (e < E) atomicAdd(deg + row[e], w[e]);
}

// ---------------------------------------------------------------------------
// norm[e] = -dinv[row]*w*dinv[col],  dinv = deg>0 ? rsqrt(max(deg,1e-30)) : 0
// ---------------------------------------------------------------------------
__global__ void norm_kernel(const int* __restrict__ row, const int* __restrict__ col,
                            const float* __restrict__ w, const float* __restrict__ deg,
                            float* __restrict__ nrm, int E) {
    int e = blockIdx.x * blockDim.x + threadIdx.x;
    if (e >= E) return;
    float dr = deg[row[e]];
    float dc = deg[col[e]];
    float ir = dr > 0.0f ? rsqrtf(fmaxf(dr, 1e-30f)) : 0.0f;
    float ic = dc > 0.0f ? rsqrtf(fmaxf(dc, 1e-30f)) : 0.0f;
    nrm[e] = -ir * w[e] * ic;
}

// ---------------------------------------------------------------------------
// SpMM scatter: Y[col[e], :] += norm[e] * X[row[e], :]
// One thread per (edge, 4-feature group); for F=128 a full wave32 covers one
// edge -> row gather is one coalesced 512B burst (L2-resident), atomics to L2.
// ---------------------------------------------------------------------------
template <int F>
__global__ void spmm_kernel(const float* __restrict__ X, const float* __restrict__ nrm,
                            const int* __restrict__ row, const int* __restrict__ col,
                            float* __restrict__ Y, int E) {
    constexpr int TPE = F / 4;  // threads per edge
    size_t gid = (size_t)blockIdx.x * blockDim.x + threadIdx.x;
    size_t total = (size_t)E * TPE;
    if (gid >= total) return;
    int e = (int)(gid / TPE);
    int f = (int)(gid % TPE) * 4;
    float w = nrm[e];
    if (w == 0.0f) return;
    int r = row[e];
    int c = col[e];
    const float4 v = *reinterpret_cast<const float4*>(X + (size_t)r * F + f);
    float* y = Y + (size_t)c * F + f;
    atomicAdd(y + 0, w * v.x);
    atomicAdd(y + 1, w * v.y);
    atomicAdd(y + 2, w * v.z);
    atomicAdd(y + 3, w * v.w);
}

// ---------------------------------------------------------------------------
// Fused Chebyshev dense stage (WMMA f32 16x16x4):
//   Out = T0*W[0] + P1*W[1] + (2*P2 - T0)*W[2] + bias   [, ReLU]
// One block = one 16-row node strip; wave w owns output-column tile w.
// A-tiles staged in LDS (stride F+1 -> conflict-free column reads).
// B loads are clamped in-bounds + masked by multiply (branch-free, no EXEC
// juggling inside the WMMA loop). Three independent accumulator chains.
// ---------------------------------------------------------------------------
template <int F, int FOUT, bool RELU>
__global__ __launch_bounds__(128) void cheb_gemm_kernel(
    const float* __restrict__ T0, const float* __restrict__ P1,
    const float* __restrict__ P2, const float* __restrict__ W,
    const float* __restrict__ bias, float* __restrict__ Out, int N) {
    __shared__ float As[3][16][F + 1];

    const int row0 = blockIdx.x * 16;

    // Cooperative stage of the three 16xF A-tiles (Tx2 computed on the fly).
    for (int idx = threadIdx.x; idx < 16 * F; idx += blockDim.x) {
        int m = idx / F;
        int k = idx - m * F;
        size_t g = (size_t)(row0 + m) * F + k;
        bool valid = (row0 + m) < N;
        float xv = valid ? T0[g] : 0.0f;
        float p1 = valid ? P1[g] : 0.0f;
        float p2 = valid ? P2[g] : 0.0f;
        As[0][m][k] = xv;
        As[1][m][k] = p1;
        As[2][m][k] = 2.0f * p2 - xv;
    }
    __syncthreads();

    const int lane = threadIdx.x & 31;
    const int n0 = (threadIdx.x >> 5) * 16;  // this wave's output-column tile
    const int m = lane & 15;                 // A row for this lane
    const int khalf = (lane < 16) ? 0 : 2;   // A/B K sub-offset per half-wave
    const int ncol = n0 + (lane & 15);       // B/D column for this lane
    const bool colok = ncol < FOUT;
    const int ncc = colok ? ncol : (FOUT - 1);  // clamped, always in bounds
    const float msk = colok ? 1.0f : 0.0f;      // branch-free OOB zeroing

    const float* __restrict__ w0 = W + (size_t)khalf * FOUT + ncc;
    const float* __restrict__ w1 = w0 + (size_t)F * FOUT;
    const float* __restrict__ w2 = w1 + (size_t)F * FOUT;

    v8f c0 = {}, c1 = {}, c2 = {};

#pragma unroll 4
    for (int k0 = 0; k0 < F; k0 += 4) {
        const int ka = k0 + khalf;
        v2f a0, a1, a2, b0, b1, b2;
        // A 16x4 f32 fragment: lane<16 -> (M=lane, K=k0..k0+1), lane>=16 -> K=k0+2..k0+3
        a0.x = As[0][m][ka];
        a0.y = As[0][m][ka + 1];
        a1.x = As[1][m][ka];
        a1.y = As[1][m][ka + 1];
        a2.x = As[2][m][ka];
        a2.y = As[2][m][ka + 1];
        // B 4x16 f32 fragment: VGPR0 = row K=ka, VGPR1 = row K=ka+1 (per half-wave)
        b0.x = w0[0] * msk;
        b0.y = w0[FOUT] * msk;
        b1.x = w1[0] * msk;
        b1.y = w1[FOUT] * msk;
        b2.x = w2[0] * msk;
        b2.y = w2[FOUT] * msk;
        w0 += 4 * FOUT;
        w1 += 4 * FOUT;
        w2 += 4 * FOUT;

        c0 = __builtin_amdgcn_wmma_f32_16x16x4_f32(false, a0, false, b0, (short)0, c0,
                                                   false, false);
        c1 = __builtin_amdgcn_wmma_f32_16x16x4_f32(false, a1, false, b1, (short)0, c1,
                                                   false, false);
        c2 = __builtin_amdgcn_wmma_f32_16x16x4_f32(false, a2, false, b2, (short)0, c2,
                                                   false, false);
    }

    // D 16x16 f32 layout: VGPR v -> M = v + 8*(lane>=16), N = lane&15
    const float bv = bias[ncc];
    const int mbase = (lane < 16) ? 0 : 8;
#pragma unroll
    for (int v = 0; v < 8; ++v) {
        int mrow = row0 + v + mbase;
        if (colok && mrow < N) {
            float val = c0[v] + c1[v] + c2[v] + bv;
            if (RELU) val = fmaxf(val, 0.0f);
            Out[(size_t)mrow * FOUT + ncol] = val;
        }
    }
}

// ---------------------------------------------------------------------------
// Orchestration
// ---------------------------------------------------------------------------
extern "C" void kernel_launch(void* const* d_in, const int* in_sizes, int n_in,
                              void* d_out, int out_size, void* d_ws, size_t ws_size,
                              hipStream_t stream) {
    (void)n_in;
    (void)out_size;
    (void)ws_size;

    constexpr int FIN = 128, FHID = 64, FOUT = 40;

    const float* x = (const float*)d_in[0];
    const int* eidx = (const int*)d_in[1];
    const float* ew = (const float*)d_in[2];
    const float* W1 = (const float*)d_in[3];
    const float* b1 = (const float*)d_in[4];
    const float* W2 = (const float*)d_in[5];
    const float* b2 = (const float*)d_in[6];
    float* out = (float*)d_out;

    const int E = in_sizes[2];
    const int N = in_sizes[0] / FIN;
    const int* row = eidx;      // edge_index[0]
    const int* col = eidx + E;  // edge_index[1]

    float* ws = (float*)d_ws;
    float* deg = ws;                   // N
    float* nrm = deg + N;              // E
    float* P1 = nrm + E;               // N*FIN (reused as N*FHID in layer 2)
    float* P2 = P1 + (size_t)N * FIN;  // N*FIN (reused as N*FHID in layer 2)
    float* h = P2 + (size_t)N * FIN;   // N*FHID

    const int T = 256;
    const int eb = (E + T - 1) / T;

    // --- normalized Laplacian edge weights ---
    zero_kernel<<<1024, T, 0, stream>>>(deg, (size_t)N);
    deg_kernel<<<eb, T, 0, stream>>>(row, ew, deg, E);
    norm_kernel<<<eb, T, 0, stream>>>(row, col, ew, deg, nrm, E);

    // --- layer 1: props (F=128) ---
    {
        size_t tot = (size_t)E * (FIN / 4);
        int pb = (int)((tot + T - 1) / T);
        zero_kernel<<<4096, T, 0, stream>>>(P1, (size_t)N * FIN);
        spmm_kernel<FIN><<<pb, T, 0, stream>>>(x, nrm, row, col, P1, E);
        zero_kernel<<<4096, T, 0, stream>>>(P2, (size_t)N * FIN);
        spmm_kernel<FIN><<<pb, T, 0, stream>>>(P1, nrm, row, col, P2, E);
    }
    // --- layer 1: fused Chebyshev GEMM + bias + ReLU -> h (N x 64) ---
    cheb_gemm_kernel<FIN, FHID, true><<<(N + 15) / 16, 4 * 32, 0, stream>>>(
        x, P1, P2, W1, b1, h, N);

    // --- layer 2: props (F=64), reuse P1/P2 ---
    {
        size_t tot = (size_t)E * (FHID / 4);
        int pb = (int)((tot + T - 1) / T);
        zero_kernel<<<2048, T, 0, stream>>>(P1, (size_t)N * FHID);
        spmm_kernel<FHID><<<pb, T, 0, stream>>>(h, nrm, row, col, P1, E);
        zero_kernel<<<2048, T, 0, stream>>>(P2, (size_t)N * FHID);
        spmm_kernel<FHID><<<pb, T, 0, stream>>>(P1, nrm, row, col, P2, E);
    }
    // --- layer 2: fused Chebyshev GEMM + bias -> out (N x 40), 3 col tiles ---
    cheb_gemm_kernel<FHID, FOUT, false><<<(N + 15) / 16, 3 * 32, 0, stream>>>(
        h, P1, P2, W2, b2, out, N);
}